// LocalizeAttention_41274635715204
// MI455X (gfx1250) — compile-verified
//
#include <hip/hip_runtime.h>
#include <stdint.h>

// LocalizeAttention: out[bh, p, f, c] = x[bh, neighbor(p, f), c] (zero if OOB).
// Pure gather, 0 FLOPs. 764 MB out / 28 MB in -> store-BW bound (~34us @ 23.3TB/s).
// CDNA5 path used: async global->LDS data mover (ASYNCcnt) + NT streaming stores.

#define GRID3   24
#define NPOS    (GRID3 * GRID3 * GRID3)      // 13824
#define DCH     32                            // channels (fp32)
#define NBH     16                            // b*h = 2*8
#define TILE    4
#define TPA     (GRID3 / TILE)                // 6 tiles per axis
#define NTILE   (TPA * TPA * TPA)             // 216 tiles per bh
#define HALO    (TILE + 2)                    // 6
#define HROWS   (HALO * HALO * HALO)          // 216 halo rows
#define HV4     (HROWS * (DCH / 4))           // 1728 float4 slots in LDS
#define OROWS   (TILE * TILE * TILE * 27)     // 1728 output rows per tile

// Native clang vector types (HIP's float4 is a struct; builtins need vectors).
typedef float v4f __attribute__((ext_vector_type(4)));
typedef int   v4i __attribute__((ext_vector_type(4)));
typedef __attribute__((address_space(1))) v4i glob_v4i;
typedef __attribute__((address_space(3))) v4i lds_v4i;

__device__ __forceinline__ void async_copy_b128(const float* src, v4f* lds_dst) {
#if __has_builtin(__builtin_amdgcn_global_load_async_to_lds_b128)
  // signature: (global int4*, local int4*, imm offset, imm cpol)
  __builtin_amdgcn_global_load_async_to_lds_b128(
      (glob_v4i*)(uintptr_t)src, (lds_v4i*)lds_dst, 0, 0);
#else
  unsigned laddr = (unsigned)(uintptr_t)(lds_v4i*)lds_dst;
  asm volatile("global_load_async_to_lds_b128 %0, %1, off"
               :: "v"(laddr), "v"(src)
               : "memory");
#endif
}

__device__ __forceinline__ void wait_async_zero() {
#if __has_builtin(__builtin_amdgcn_s_wait_asynccnt)
  __builtin_amdgcn_s_wait_asynccnt(0);
#else
  asm volatile("s_wait_asynccnt 0" ::: "memory");
#endif
}

__global__ __launch_bounds__(256) void localize_attn_kernel(
    const float* __restrict__ x, float* __restrict__ out) {
  __shared__ v4f tile[HV4];  // 27,648 bytes

  const int tid = threadIdx.x;
  const int tile_id = blockIdx.x;
  const int bh = tile_id / NTILE;
  const int t  = tile_id % NTILE;
  const int ti = (t / (TPA * TPA)) * TILE;
  const int tj = ((t / TPA) % TPA) * TILE;
  const int tk = (t % TPA) * TILE;

  const float* xb = x + (size_t)bh * NPOS * DCH;

  // ---- Stage 6x6x6 halo rows into LDS via async data mover -----------------
  for (int idx = tid; idx < HV4; idx += 256) {
    const int row = idx >> 3;          // halo row 0..215
    const int sub = idx & 7;           // float4 within the 32-ch row
    const int hi = row / (HALO * HALO);
    const int hj = (row / HALO) % HALO;
    const int hk = row % HALO;
    // Clamp OOB halo coords; clamped (garbage) entries are never read back
    // because validity is re-checked at store time.
    int gi = ti + hi - 1; gi = gi < 0 ? 0 : (gi > GRID3 - 1 ? GRID3 - 1 : gi);
    int gj = tj + hj - 1; gj = gj < 0 ? 0 : (gj > GRID3 - 1 ? GRID3 - 1 : gj);
    int gk = tk + hk - 1; gk = gk < 0 ? 0 : (gk > GRID3 - 1 ? GRID3 - 1 : gk);
    const float* src = xb + ((size_t)((gi * GRID3 + gj) * GRID3 + gk) * DCH + sub * 4);
    async_copy_b128(src, &tile[idx]);
  }
  wait_async_zero();   // drain this wave's ASYNCcnt before publishing LDS
  __syncthreads();     // s_barrier_signal/wait -1: all waves' tiles visible

  // ---- Emit 64 positions x 27 filters, 8 lanes -> one 128B row -------------
  float* ob = out + (size_t)bh * NPOS * 27 * DCH;
  const int sub = tid & 7;
  for (int r = tid >> 3; r < OROWS; r += 32) {
    const int p = r / 27;              // local position 0..63
    const int f = r % 27;              // filter index fi*9+fj*3+fk
    const int li = p >> 4, lj = (p >> 2) & 3, lk = p & 3;
    const int fi = f / 9, fj = (f / 3) % 3, fk = f % 3;

    const int gi = ti + li + fi - 1;
    const int gj = tj + lj + fj - 1;
    const int gk = tk + lk + fk - 1;
    const bool ok = ((unsigned)gi < (unsigned)GRID3) &
                    ((unsigned)gj < (unsigned)GRID3) &
                    ((unsigned)gk < (unsigned)GRID3);

    // halo-local coords of the neighbor: (li+fi, lj+fj, lk+fk) in [0,6)
    const int hrow = ((li + fi) * HALO + (lj + fj)) * HALO + (lk + fk);
    v4f v = 0.0f;
    if (ok) v = tile[hrow * 8 + sub];

    const int pg = ((ti + li) * GRID3 + (tj + lj)) * GRID3 + (tk + lk);
    v4f* dst = (v4f*)(ob + ((size_t)pg * 27 + f) * DCH) + sub;
    __builtin_nontemporal_store(v, dst);  // TH=NT: don't pollute L2 with output
  }
}

extern "C" void kernel_launch(void* const* d_in, const int* in_sizes, int n_in,
                              void* d_out, int out_size, void* d_ws, size_t ws_size,
                              hipStream_t stream) {
  (void)in_sizes; (void)n_in; (void)d_ws; (void)ws_size; (void)out_size;
  const float* x = (const float*)d_in[0];
  float* out = (float*)d_out;
  const int blocks = NBH * NTILE;  // 3456
  localize_attn_kernel<<<blocks, 256, 0, stream>>>(x, out);
}